// NetVLAD_V1_53472342835390
// MI455X (gfx1250) — compile-verified
//
#include <hip/hip_runtime.h>
#include <math.h>

// Problem constants (from reference)
#define NN 4
#define CC 128
#define PP 784     // H*W = 28*28
#define DD 512
#define KK 64
#define EPSN 1e-12f

typedef __attribute__((ext_vector_type(2))) float v2f;
typedef __attribute__((ext_vector_type(8))) float v8f;

// ---------------------------------------------------------------------------
// Kernel 0: per-pixel inverse L2 norm over channels.
// invn[n*PP+p] = 1 / max(||x[n,:,p]||_2, EPS)
// Consecutive threads -> consecutive p -> fully coalesced loads per c-step.
// ---------------------------------------------------------------------------
__global__ void __launch_bounds__(256)
netvlad_invnorm(const float* __restrict__ x, float* __restrict__ invn) {
  int idx = blockIdx.x * blockDim.x + threadIdx.x;
  if (idx >= NN * PP) return;
  int n = idx / PP, p = idx - n * PP;
  const float* xp = x + (size_t)n * CC * PP + p;
  float ss = 0.0f;
#pragma unroll 8
  for (int c = 0; c < CC; ++c) {
    float v = xp[(size_t)c * PP];
    ss = fmaf(v, v, ss);
  }
  invn[idx] = 1.0f / fmaxf(sqrtf(ss), EPSN);
}

// ---------------------------------------------------------------------------
// Kernel 1: fused normalize + 1x1 conv as WMMA f32 GEMM.
// feat[n][d][p] = sum_c (x[n,c,p]*invn[n,p]) * w[d,c] + bias[d]
// One wave per 16(p) x 16(d) tile; K-loop over C in steps of 4 using
// V_WMMA_F32_16X16X4_F32.
//   A (16x4, M=p,K=c) f32 layout: lanes 0-15: M=lane, v0=K0 v1=K1;
//                                 lanes 16-31: M=lane-16, v0=K2 v1=K3.
//   B (4x16, K=c,N=d) mirrored:   lanes 0-15: N=lane, v0=K0 v1=K1;
//                                 lanes 16-31: N=lane-16, v0=K2 v1=K3.
//   C/D (16x16) f32 layout: lane<16 -> (M=r, N=lane); lane>=16 -> (M=r+8).
// ---------------------------------------------------------------------------
__global__ void __launch_bounds__(256)
netvlad_gemm(const float* __restrict__ x, const float* __restrict__ w,
             const float* __restrict__ bias, const float* __restrict__ invn,
             float* __restrict__ feat) {
  const int lane = threadIdx.x & 31;
  const int wid  = blockIdx.x * (blockDim.x >> 5) + (threadIdx.x >> 5);
  // wid in [0, 4*49*32): dt fastest, then pt, then n
  const int dt = wid & 31;             // d-tile: 512/16 = 32
  const int pt = (wid >> 5) % 49;      // p-tile: 784/16 = 49
  const int n  = wid / (32 * 49);

  const int hl   = lane & 15;
  const int hi   = lane >> 4;          // 0: K0/K1 half, 1: K2/K3 half
  const int prow = pt * 16 + hl;       // pixel row owned by this lane (A side)
  const int dcol = dt * 16 + hl;       // d column owned by this lane (B/C side)

  const float inv = invn[n * PP + prow];          // loop-invariant per lane
  const float* xb = x + (size_t)n * CC * PP + prow;
  const float* wb = w + (size_t)dcol * CC;

  v8f acc = {};
#pragma unroll 4
  for (int c0 = 0; c0 < CC; c0 += 4) {
    const int ca = c0 + (hi << 1);     // c0 / c0+2 depending on lane half
    v2f a, b;
    a.x = xb[(size_t)(ca + 0) * PP] * inv;
    a.y = xb[(size_t)(ca + 1) * PP] * inv;
    b = *(const v2f*)(wb + ca);        // w[dcol][ca], w[dcol][ca+1] (8B aligned)
    acc = __builtin_amdgcn_wmma_f32_16x16x4_f32(
        /*neg_a=*/false, a, /*neg_b=*/false, b,
        /*c_mod=*/(short)0, acc, /*reuse_a=*/false, /*reuse_b=*/false);
  }

  const float bv = bias[dcol];
  // Each lane holds 8 consecutive p values for a single d column.
  float* out = feat + ((size_t)(n * DD + dcol)) * PP + pt * 16 + (hi << 3);
#pragma unroll
  for (int r = 0; r < 8; ++r) out[r] = acc[r] + bv;
}

// ---------------------------------------------------------------------------
// Kernel 2: per-(n,d) softmax over p (shift-invariant => centroids drop out),
// S[n,d] = sum_p softmax_p(feat) * feat; broadcast to all K clusters.
// One wave per row; 784 elements -> up to 25 per lane, kept in registers.
// ---------------------------------------------------------------------------
__global__ void __launch_bounds__(256)
netvlad_softmax_vlad(const float* __restrict__ feat, float* __restrict__ out) {
  const int lane = threadIdx.x & 31;
  const int row  = blockIdx.x * (blockDim.x >> 5) + (threadIdx.x >> 5); // n*DD+d
  const int n = row / DD, d = row - n * DD;
  const float* f = feat + (size_t)row * PP;

  float vals[25];
  int cnt = 0;
#pragma unroll
  for (int i = 0; i < 25; ++i) {
    int p = lane + (i << 5);
    if (p < PP) { vals[i] = f[p]; cnt = i + 1; }
  }

  float m = -INFINITY;
  for (int i = 0; i < cnt; ++i) m = fmaxf(m, vals[i]);
  for (int off = 16; off > 0; off >>= 1) m = fmaxf(m, __shfl_xor(m, off, 32));

  float s1 = 0.0f, s2 = 0.0f;
  for (int i = 0; i < cnt; ++i) {
    float e = expf(vals[i] - m);
    s1 += e;
    s2 = fmaf(e, vals[i], s2);
  }
  for (int off = 16; off > 0; off >>= 1) {
    s1 += __shfl_xor(s1, off, 32);
    s2 += __shfl_xor(s2, off, 32);
  }
  const float S = s2 / s1;

  // vlad[n][k][d] = S for all k; each lane writes k=lane and k=lane+32.
  out[((size_t)n * KK + lane) * DD + d]        = S;
  out[((size_t)n * KK + lane + 32) * DD + d]   = S;
}

// ---------------------------------------------------------------------------
extern "C" void kernel_launch(void* const* d_in, const int* in_sizes, int n_in,
                              void* d_out, int out_size, void* d_ws, size_t ws_size,
                              hipStream_t stream) {
  const float* x    = (const float*)d_in[0];  // (N,C,H,W)
  const float* w    = (const float*)d_in[1];  // (D,C)
  const float* b    = (const float*)d_in[2];  // (D,)
  // d_in[3] (centroids) is mathematically unused: softmax over p is
  // shift-invariant, so the centroid subtraction cancels exactly.
  (void)in_sizes; (void)n_in; (void)out_size; (void)ws_size;

  float* ws   = (float*)d_ws;
  float* invn = ws;            // N*PP = 3136 floats
  float* feat = ws + 4096;     // N*DD*PP = 1,605,632 floats (~6.4 MB)
  float* out  = (float*)d_out; // (N,K,D) = 131072 floats

  netvlad_invnorm<<<(NN * PP + 255) / 256, 256, 0, stream>>>(x, invn);
  // 4 * 49 * 32 = 6272 tiles/waves; 8 waves per 256-thread block -> 784 blocks
  netvlad_gemm<<<784, 256, 0, stream>>>(x, w, b, invn, feat);
  // N*DD = 2048 rows, 8 waves per block -> 256 blocks
  netvlad_softmax_vlad<<<(NN * DD) / 8, 256, 0, stream>>>(feat, out);
}